// VectorQuantizer_46540265620156
// MI455X (gfx1250) — compile-verified
//
#include <hip/hip_runtime.h>
#include <hip/hip_bf16.h>

// ---------------------------------------------------------------------------
// VQ-VAE vector quantizer for gfx1250 (MI455X).
//   z:        [64, 64, 32, 32] f32   (B, D, H, W)  -> flat_z [N=65536, D=64]
//   codebook: [1024, 64] f32         (K, D)
// Outputs (concatenated in d_out, f32):
//   quantized_st [64,64,32,32] (4194304) | vq_loss (1) | perplexity (1)
//   | utilization (1) | encoding_indices (65536, stored as float)
// Distance GEMM on V_WMMA_F32_16X16X4_F32 (full f32 matrix pipe), codebook
// resident bank-conflict-free in the 320KB WGP LDS, dual accumulator chains
// for back-to-back WMMA issue, native v_min_u64 packed argmin.
// ---------------------------------------------------------------------------

typedef __attribute__((ext_vector_type(2))) float v2f;
typedef __attribute__((ext_vector_type(8))) float v8f;

#define VQ_N        65536      // B*H*W
#define VQ_K        1024
#define VQ_D        64
#define VQ_HW       1024       // H*W
#define VQ_TILES    4096       // N / 16
#define KSTRIDE     68         // 64 data + 4 pad floats -> conflict-free LDS

#define BLOCKS      256
#define THREADS     512        // 16 wave32 waves -> 256*16 = 4096 waves, 1 tile each

__device__ __forceinline__ unsigned long long vq_shflx64(unsigned long long v, int m) {
  unsigned lo = (unsigned)__shfl_xor((int)(unsigned)(v & 0xFFFFFFFFULL), m, 32);
  unsigned hi = (unsigned)__shfl_xor((int)(unsigned)(v >> 32), m, 32);
  return ((unsigned long long)hi << 32) | (unsigned long long)lo;
}

__global__ void vq_init_kernel(unsigned int* __restrict__ counts) {
  counts[threadIdx.x] = 0u;   // <<<1,1024>>>
}

__launch_bounds__(THREADS, 1)
__global__ void vq_argmin_kernel(const float* __restrict__ z,
                                 const float* __restrict__ cb,
                                 float* __restrict__ q_out,
                                 float* __restrict__ enc_out,
                                 unsigned int* __restrict__ counts,
                                 float* __restrict__ loss_part) {
  __shared__ float scb[VQ_K * KSTRIDE];   // 272 KB codebook (padded)
  __shared__ float snorm[VQ_K];           //   4 KB ||e||^2

  const int tid = threadIdx.x;

  // ---- stage codebook into LDS (coalesced global reads) ----
  for (int i = tid; i < VQ_K * VQ_D; i += THREADS) {
    scb[(i >> 6) * KSTRIDE + (i & 63)] = cb[i];
  }
  __syncthreads();

  // ---- per-entry squared norms ----
  for (int e = tid; e < VQ_K; e += THREADS) {
    float s = 0.0f;
    const float* row = &scb[e * KSTRIDE];
    #pragma unroll
    for (int i = 0; i < VQ_D; ++i) s = fmaf(row[i], row[i], s);
    snorm[e] = s;
  }
  __syncthreads();

  // ---- each wave handles one 16-row tile of flat_z ----
  const int wave    = tid >> 5;
  const int lane    = tid & 31;
  const int rowlane = lane & 15;          // z row within tile / codebook row in group
  const int hi      = lane >> 4;          // 0: K pair {0,1}, 1: K pair {2,3}
  const int tile    = blockIdx.x * (THREADS / 32) + wave;   // 0..4095
  const int n0      = tile * 16;
  const int b       = n0 >> 10;           // image index (H*W = 1024)
  const int s0      = n0 & 1023;          // spatial offset within image

  // Load A: z-tile 16x64 f32 in WMMA A layout. Lane holds 32 floats:
  //   A[2*kk+j] = flat_z[n0+rowlane][4*kk + 2*hi + j]
  float a[32];
  const float* zb = z + (size_t)b * (VQ_D * VQ_HW) + s0 + rowlane;
  #pragma unroll
  for (int kk = 0; kk < 16; ++kk) {
    const int c0 = kk * 4 + hi * 2;
    a[2 * kk + 0] = zb[(size_t)(c0 + 0) * VQ_HW];
    a[2 * kk + 1] = zb[(size_t)(c0 + 1) * VQ_HW];
  }

  // Running argmin over all K. Key = monotonic(f32 bits of ||e||^2 - 2 z.e)
  // packed with the codebook index (||z||^2 is row-constant -> dropped).
  unsigned long long best[8];
  #pragma unroll
  for (int r = 0; r < 8; ++r) best[r] = ~0ULL;

  for (int g = 0; g < 64; ++g) {
    const int cbrow = g * 16 + rowlane;
    const float cn = snorm[cbrow];
    v8f c0 = {};   // even k-chunks
    v8f c1 = {};   // odd k-chunks -> two independent WMMA dependency chains
    #pragma unroll
    for (int kk = 0; kk < 16; kk += 2) {
      v2f av0 = {a[2 * kk + 0], a[2 * kk + 1]};
      v2f av1 = {a[2 * kk + 2], a[2 * kk + 3]};
      v2f bv0 = *(const v2f*)&scb[cbrow * KSTRIDE + (kk + 0) * 4 + hi * 2];
      v2f bv1 = *(const v2f*)&scb[cbrow * KSTRIDE + (kk + 1) * 4 + hi * 2];
      c0 = __builtin_amdgcn_wmma_f32_16x16x4_f32(
          false, av0, false, bv0, (short)0, c0, false, false);
      c1 = __builtin_amdgcn_wmma_f32_16x16x4_f32(
          false, av1, false, bv1, (short)0, c1, false, false);
    }
    #pragma unroll
    for (int r = 0; r < 8; ++r) {
      // C[r] @ lane: element (m = r + 8*hi, n = cbrow)
      const float d = fmaf(-2.0f, c0[r] + c1[r], cn);
      unsigned int bits = __float_as_uint(d);
      bits ^= (bits & 0x80000000u) ? 0xFFFFFFFFu : 0x80000000u;
      const unsigned long long key =
          ((unsigned long long)bits << 32) | (unsigned long long)(unsigned)cbrow;
      if (key < best[r]) best[r] = key;
    }
  }

  // Cross-lane min within each 16-lane half (half selects m vs m+8).
  #pragma unroll
  for (int r = 0; r < 8; ++r) {
    unsigned long long k = best[r];
    #pragma unroll
    for (int off = 1; off < 16; off <<= 1) {
      const unsigned long long o = vq_shflx64(k, off);
      if (o < k) k = o;
    }
    best[r] = k;  // lanes 0-15: argmin of row r; lanes 16-31: row r+8
  }

  // Distribute: lane needs enc of row (lane & 15).
  int myenc = 0;
  #pragma unroll
  for (int r = 0; r < 8; ++r) {
    const int e_lo = __shfl((int)(unsigned)(best[r] & 0xFFFFFFFFULL), r, 32);
    const int e_hi = __shfl((int)(unsigned)(best[r] & 0xFFFFFFFFULL), 16 + r, 32);
    if ((rowlane & 7) == r) myenc = (rowlane & 8) ? e_hi : e_lo;
  }

  // Indices + histogram (lanes 0-15 own rows 0-15 once).
  if (hi == 0) {
    enc_out[n0 + rowlane] = (float)myenc;
    atomicAdd(&counts[myenc], 1u);
  }

  // Quantized gather-scatter + exact per-wave loss partial.
  float lsum = 0.0f;
  const float* qrow = &scb[myenc * KSTRIDE];
  float* ob = q_out + (size_t)b * (VQ_D * VQ_HW) + s0 + rowlane;
  #pragma unroll
  for (int kk = 0; kk < 16; ++kk) {
    const int c0 = kk * 4 + hi * 2;
    const v2f q = *(const v2f*)&qrow[c0];
    const float d0 = q.x - a[2 * kk + 0];
    const float d1 = q.y - a[2 * kk + 1];
    lsum = fmaf(d0, d0, lsum);
    lsum = fmaf(d1, d1, lsum);
    ob[(size_t)(c0 + 0) * VQ_HW] = q.x;
    ob[(size_t)(c0 + 1) * VQ_HW] = q.y;
  }
  #pragma unroll
  for (int off = 16; off > 0; off >>= 1) lsum += __shfl_xor(lsum, off, 32);
  if (lane == 0) loss_part[tile] = lsum;   // deterministic: one slot per wave
}

__global__ void vq_finalize_kernel(const unsigned int* __restrict__ counts,
                                   const float* __restrict__ loss_part,
                                   float* __restrict__ scalars) {
  __shared__ float sred[1024];
  const int t = threadIdx.x;

  const unsigned c = counts[t];
  const float p    = (float)c * (1.0f / (float)VQ_N);
  const float ent  = p * logf(p + 1e-10f);
  const float util = (c != 0u) ? 1.0f : 0.0f;
  float ls = loss_part[t] + loss_part[t + 1024] +
             loss_part[t + 2048] + loss_part[t + 3072];

  float tot_ent, tot_util, tot_ls;
  sred[t] = ent; __syncthreads();
  for (int s = 512; s > 0; s >>= 1) { if (t < s) sred[t] += sred[t + s]; __syncthreads(); }
  tot_ent = sred[0]; __syncthreads();

  sred[t] = util; __syncthreads();
  for (int s = 512; s > 0; s >>= 1) { if (t < s) sred[t] += sred[t + s]; __syncthreads(); }
  tot_util = sred[0]; __syncthreads();

  sred[t] = ls; __syncthreads();
  for (int s = 512; s > 0; s >>= 1) { if (t < s) sred[t] += sred[t + s]; __syncthreads(); }
  tot_ls = sred[0];

  if (t == 0) {
    // codebook_loss == commitment_loss numerically -> (1 + 0.25) * MSE
    scalars[0] = tot_ls * (1.25f / (float)(VQ_N * VQ_D));
    scalars[1] = expf(-tot_ent);                  // perplexity
    scalars[2] = tot_util * (1.0f / (float)VQ_K); // utilization
  }
}

extern "C" void kernel_launch(void* const* d_in, const int* in_sizes, int n_in,
                              void* d_out, int out_size, void* d_ws, size_t ws_size,
                              hipStream_t stream) {
  (void)in_sizes; (void)n_in; (void)out_size; (void)ws_size;
  const float* z  = (const float*)d_in[0];
  const float* cb = (const float*)d_in[1];

  float* out     = (float*)d_out;
  float* q_out   = out;                       // 4194304 floats
  float* scalars = out + (size_t)VQ_N * VQ_D; // vq_loss, perplexity, utilization
  float* enc_out = scalars + 3;               // 65536 floats

  unsigned int* counts  = (unsigned int*)d_ws;        // 1024 u32
  float* loss_part      = (float*)d_ws + VQ_K;        // 4096 f32 wave partials

  vq_init_kernel<<<1, 1024, 0, stream>>>(counts);
  vq_argmin_kernel<<<BLOCKS, THREADS, 0, stream>>>(z, cb, q_out, enc_out, counts, loss_part);
  vq_finalize_kernel<<<1, 1024, 0, stream>>>(counts, loss_part, scalars);
}